// GNNAgent_74663711474035
// MI455X (gfx1250) — compile-verified
//
#include <hip/hip_runtime.h>
#include <math.h>

typedef __attribute__((ext_vector_type(2))) float v2f;
typedef __attribute__((ext_vector_type(8))) float v8f;

#define NNODES 100000
#define NEDGES 3200000
#define NF 16
#define EF 8
#define NTILES (NNODES / 16)          // 6250, N divisible by 16

// workspace layout in floats
#define WS_AGG   0
#define WS_H1    (NNODES * NF)               // 1,600,000
#define WS_H2    (WS_H1 + NNODES * 32)       // 4,800,000
#define WS_STATS (WS_H2 + NNODES * 32)       // 8,000,000  (512 floats)
// per-layer stats block (128 floats): [0,32) sum, [32,64) sumsq,
// [64,96) scale = g*rsqrt(var+eps), [96,128) shift = beta - mean*scale

__device__ __forceinline__ v8f wmma_f32_k4(v2f a, v2f b, v8f c) {
    // D = A(16x4) * B(4x16) + C, fp32 WMMA (CDNA5)
    return __builtin_amdgcn_wmma_f32_16x16x4_f32(
        /*neg_a=*/false, a, /*neg_b=*/false, b,
        /*c_mod=*/(short)0, c, /*reuse_a=*/false, /*reuse_b=*/false);
}

__device__ __forceinline__ float gelu_exact(float v) {
    return 0.5f * v * (1.0f + erff(v * 0.70710678118654752f));
}

// ---------------- init: zero agg + stats ----------------
__global__ void init_zero_kernel(float* __restrict__ agg, float* __restrict__ stats) {
    int i = blockIdx.x * blockDim.x + threadIdx.x;
    if (i < NNODES * NF) agg[i] = 0.0f;
    if (i < 512) stats[i] = 0.0f;
}

// ---------------- edge phase: e = ea@We+be; msg = relu(x[src]+e); atomic scatter to agg[dst]
__global__ __launch_bounds__(256)
void edge_kernel(const float* __restrict__ x, const int* __restrict__ ei,
                 const float* __restrict__ ea, const float* __restrict__ We,
                 const float* __restrict__ be, float* __restrict__ agg) {
    __shared__ float sWe[EF * NF];
    __shared__ float sbe[NF];
    int t = threadIdx.x;
    if (t < EF * NF) sWe[t] = We[t];
    if (t < NF) sbe[t] = be[t];
    __syncthreads();

    int e = blockIdx.x * blockDim.x + t;
    if (e >= NEDGES) return;

    int src = ei[e];
    int dst = ei[NEDGES + e];

    const float4* eap = (const float4*)(ea + (size_t)e * EF);
    float4 a0 = eap[0], a1 = eap[1];
    float av[EF] = {a0.x, a0.y, a0.z, a0.w, a1.x, a1.y, a1.z, a1.w};

    const float* xr = x + (size_t)src * NF;
    float* ag = agg + (size_t)dst * NF;
#pragma unroll
    for (int f = 0; f < NF; ++f) {
        float s = sbe[f];
#pragma unroll
        for (int k = 0; k < EF; ++k) s += av[k] * sWe[k * NF + f];
        s += xr[f];
        s = s > 0.0f ? s : 0.0f;
        unsafeAtomicAdd(&ag[f], s);   // global_atomic_add_f32, L2-resident
    }
}

// ---------------- GEMM1: h1 = ((1+eps)*x + agg) @ W1 + b1, + BN1 partial sums
__global__ __launch_bounds__(256)
void gemm1_kernel(const float* __restrict__ x, const float* __restrict__ agg,
                  const float* __restrict__ W1, const float* __restrict__ b1,
                  const float* __restrict__ eps_p,
                  float* __restrict__ h1, float* __restrict__ stats1) {
    int lane = threadIdx.x & 31;
    int wave = threadIdx.x >> 5;
    int tile = blockIdx.x * (blockDim.x >> 5) + wave;
    if (tile >= NTILES) return;                 // wave-uniform: EXEC stays all-1s

    float epsv = 1.0f + eps_p[0];
    int row = tile * 16 + (lane & 15);
    int kb = (lane >> 4) * 2;                   // lanes 16..31 carry K=+2,+3

    v2f afrag[4];
#pragma unroll
    for (int s = 0; s < 4; ++s) {
        int k = 4 * s + kb;
        float2 xv = *(const float2*)(x + (size_t)row * NF + k);
        float2 av = *(const float2*)(agg + (size_t)row * NF + k);
        afrag[s].x = epsv * xv.x + av.x;
        afrag[s].y = epsv * xv.y + av.y;
    }

    int ncol = lane & 15;
    int mbase = tile * 16 + ((lane >> 4) * 8);
#pragma unroll
    for (int nt = 0; nt < 2; ++nt) {
        int n = nt * 16 + ncol;
        v8f c = {};
#pragma unroll
        for (int s = 0; s < 4; ++s) {
            int k = 4 * s + kb;
            v2f b;
            b.x = W1[k * 32 + n];
            b.y = W1[(k + 1) * 32 + n];
            c = wmma_f32_k4(afrag[s], b, c);
        }
        float bias = b1[n];
        float sum = 0.0f, sq = 0.0f;
#pragma unroll
        for (int r = 0; r < 8; ++r) {
            float v = c[r] + bias;
            h1[(size_t)(mbase + r) * 32 + n] = v;
            sum += v; sq += v * v;
        }
        unsafeAtomicAdd(&stats1[n], sum);
        unsafeAtomicAdd(&stats1[32 + n], sq);
    }
}

// ---------------- BN stats finalize: scale/shift from sum/sumsq ----------------
__global__ void bnstats_kernel(float* __restrict__ stats, const float* __restrict__ g,
                               const float* __restrict__ bt) {
    int n = threadIdx.x;          // 32 threads
    float invN = 1.0f / (float)NNODES;
    float mean = stats[n] * invN;
    float var = stats[32 + n] * invN - mean * mean;   // biased var (ddof=0)
    float sc = g[n] * rsqrtf(var + 1e-5f);
    stats[64 + n] = sc;
    stats[96 + n] = bt[n] - mean * sc;
}

// ---------------- GEMM2: h2 = gelu(BN1(h1)) @ W2 + b2, + BN2 partial sums
__global__ __launch_bounds__(256)
void gemm2_kernel(const float* __restrict__ h1, const float* __restrict__ W2,
                  const float* __restrict__ b2, const float* __restrict__ stats1,
                  float* __restrict__ h2, float* __restrict__ stats2) {
    int lane = threadIdx.x & 31;
    int wave = threadIdx.x >> 5;
    int tile = blockIdx.x * (blockDim.x >> 5) + wave;
    if (tile >= NTILES) return;

    int row = tile * 16 + (lane & 15);
    int kb = (lane >> 4) * 2;

    v2f afrag[8];
#pragma unroll
    for (int s = 0; s < 8; ++s) {
        int k = 4 * s + kb;
        float2 hv = *(const float2*)(h1 + (size_t)row * 32 + k);
        float v0 = gelu_exact(hv.x * stats1[64 + k]     + stats1[96 + k]);
        float v1 = gelu_exact(hv.y * stats1[64 + k + 1] + stats1[96 + k + 1]);
        afrag[s].x = v0;
        afrag[s].y = v1;
    }

    int ncol = lane & 15;
    int mbase = tile * 16 + ((lane >> 4) * 8);
#pragma unroll
    for (int nt = 0; nt < 2; ++nt) {
        int n = nt * 16 + ncol;
        v8f c = {};
#pragma unroll
        for (int s = 0; s < 8; ++s) {
            int k = 4 * s + kb;
            v2f b;
            b.x = W2[k * 32 + n];
            b.y = W2[(k + 1) * 32 + n];
            c = wmma_f32_k4(afrag[s], b, c);
        }
        float bias = b2[n];
        float sum = 0.0f, sq = 0.0f;
#pragma unroll
        for (int r = 0; r < 8; ++r) {
            float v = c[r] + bias;
            h2[(size_t)(mbase + r) * 32 + n] = v;
            sum += v; sq += v * v;
        }
        unsafeAtomicAdd(&stats2[n], sum);
        unsafeAtomicAdd(&stats2[32 + n], sq);
    }
}

// ---------------- GEMM3: out = gelu(BN2(h2)) @ W3 + b3  ([N,64]) ----------------
__global__ __launch_bounds__(256)
void gemm3_kernel(const float* __restrict__ h2, const float* __restrict__ W3,
                  const float* __restrict__ b3, const float* __restrict__ stats2,
                  float* __restrict__ out) {
    int lane = threadIdx.x & 31;
    int wave = threadIdx.x >> 5;
    int tile = blockIdx.x * (blockDim.x >> 5) + wave;
    if (tile >= NTILES) return;

    int row = tile * 16 + (lane & 15);
    int kb = (lane >> 4) * 2;

    v2f afrag[8];
#pragma unroll
    for (int s = 0; s < 8; ++s) {
        int k = 4 * s + kb;
        float2 hv = *(const float2*)(h2 + (size_t)row * 32 + k);
        afrag[s].x = gelu_exact(hv.x * stats2[64 + k]     + stats2[96 + k]);
        afrag[s].y = gelu_exact(hv.y * stats2[64 + k + 1] + stats2[96 + k + 1]);
    }

    int ncol = lane & 15;
    int mbase = tile * 16 + ((lane >> 4) * 8);
#pragma unroll
    for (int nt = 0; nt < 4; ++nt) {
        int n = nt * 16 + ncol;
        v8f c = {};
#pragma unroll
        for (int s = 0; s < 8; ++s) {
            int k = 4 * s + kb;
            v2f b;
            b.x = W3[k * 64 + n];
            b.y = W3[(k + 1) * 64 + n];
            c = wmma_f32_k4(afrag[s], b, c);
        }
        float bias = b3[n];
#pragma unroll
        for (int r = 0; r < 8; ++r) {
            out[(size_t)(mbase + r) * 64 + n] = c[r] + bias;
        }
    }
}

extern "C" void kernel_launch(void* const* d_in, const int* in_sizes, int n_in,
                              void* d_out, int out_size, void* d_ws, size_t ws_size,
                              hipStream_t stream) {
    const float* x    = (const float*)d_in[0];
    const int*   ei   = (const int*)d_in[1];
    const float* ea   = (const float*)d_in[2];
    const float* We   = (const float*)d_in[3];
    const float* be   = (const float*)d_in[4];
    const float* W1   = (const float*)d_in[5];
    const float* b1   = (const float*)d_in[6];
    const float* g1   = (const float*)d_in[7];
    const float* bt1  = (const float*)d_in[8];
    const float* W2   = (const float*)d_in[9];
    const float* b2   = (const float*)d_in[10];
    const float* epsp = (const float*)d_in[11];
    const float* g2   = (const float*)d_in[12];
    const float* bt2  = (const float*)d_in[13];
    const float* W3   = (const float*)d_in[14];
    const float* b3   = (const float*)d_in[15];

    float* ws     = (float*)d_ws;
    float* agg    = ws + WS_AGG;
    float* h1     = ws + WS_H1;
    float* h2     = ws + WS_H2;
    float* stats1 = ws + WS_STATS;
    float* stats2 = ws + WS_STATS + 128;
    float* out    = (float*)d_out;

    // 1) zero agg + stats
    {
        int total = NNODES * NF;
        init_zero_kernel<<<(total + 255) / 256, 256, 0, stream>>>(agg, stats1);
    }
    // 2) edge scatter
    edge_kernel<<<(NEDGES + 255) / 256, 256, 0, stream>>>(x, ei, ea, We, be, agg);
    // 3) GEMM1 + BN1 partials (8 waves/block, 1 tile of 16 nodes per wave)
    int gemm_blocks = (NTILES + 7) / 8;
    gemm1_kernel<<<gemm_blocks, 256, 0, stream>>>(x, agg, W1, b1, epsp, h1, stats1);
    // 4) BN1 finalize
    bnstats_kernel<<<1, 32, 0, stream>>>(stats1, g1, bt1);
    // 5) BN1+GELU -> GEMM2 + BN2 partials
    gemm2_kernel<<<gemm_blocks, 256, 0, stream>>>(h1, W2, b2, stats1, h2, stats2);
    // 6) BN2 finalize
    bnstats_kernel<<<1, 32, 0, stream>>>(stats2, g2, bt2);
    // 7) BN2+GELU -> GEMM3 -> out
    gemm3_kernel<<<gemm_blocks, 256, 0, stream>>>(h2, W3, b3, stats2, out);
}